// Net_27908697490044
// MI455X (gfx1250) — compile-verified
//
#include <hip/hip_runtime.h>
#include <hip/hip_bf16.h>

// ---------------- problem constants (match reference) ----------------
#define NNODES   50000
#define NEDGES   800000
#define NFEAT    512
#define NHID     256
#define NCLS     40
#define NCLS_PAD 48   // 3 tiles of 16

typedef __attribute__((ext_vector_type(2))) float v2f;
typedef __attribute__((ext_vector_type(8))) float v8f;

// =====================================================================
// GEMM 1: h = X @ W1   [50000 x 512] * [512 x 256]
// grid.x = 50000/16 = 3125 row tiles; block = (32,4): 4 waves, each wave
// computes a 16x64 slab (4 accumulators), so one block covers all 256 cols.
// Uses V_WMMA_F32_16X16X4_F32, K-loop of 128 steps (512 wmma / wave).
// =====================================================================
__global__ __launch_bounds__(128)
void gcn_gemm1(const float* __restrict__ X,
               const float* __restrict__ W1,
               float* __restrict__ H) {
    const int lane  = threadIdx.x;            // 0..31
    const int m0    = blockIdx.x * 16;
    const int n0    = threadIdx.y * 64;       // wave's 64-col slab
    const int half  = lane >> 4;              // 0: K pair {0,1}, 1: {2,3}
    const int kh    = half * 2;
    const int l15   = lane & 15;
    const int mrow  = m0 + l15;               // A-fragment row for this lane

    v8f acc0 = {}, acc1 = {}, acc2 = {}, acc3 = {};

    const float* arow = X + (size_t)mrow * NFEAT;

    for (int k = 0; k < NFEAT; k += 4) {
        // A fragment 16x4 f32: lanes 0-15 hold K=k+0,k+1; lanes 16-31 K=k+2,k+3
        v2f a;
        a.x = arow[k + kh];
        a.y = arow[k + kh + 1];

        const float* brow0 = W1 + (size_t)(k + kh) * NHID;
        const float* brow1 = brow0 + NHID;

        v2f b;
        b.x = brow0[n0 + 0 * 16 + l15];
        b.y = brow1[n0 + 0 * 16 + l15];
        acc0 = __builtin_amdgcn_wmma_f32_16x16x4_f32(false, a, false, b, (short)0, acc0, false, false);

        b.x = brow0[n0 + 1 * 16 + l15];
        b.y = brow1[n0 + 1 * 16 + l15];
        acc1 = __builtin_amdgcn_wmma_f32_16x16x4_f32(false, a, false, b, (short)0, acc1, false, false);

        b.x = brow0[n0 + 2 * 16 + l15];
        b.y = brow1[n0 + 2 * 16 + l15];
        acc2 = __builtin_amdgcn_wmma_f32_16x16x4_f32(false, a, false, b, (short)0, acc2, false, false);

        b.x = brow0[n0 + 3 * 16 + l15];
        b.y = brow1[n0 + 3 * 16 + l15];
        acc3 = __builtin_amdgcn_wmma_f32_16x16x4_f32(false, a, false, b, (short)0, acc3, false, false);
    }

    // D layout: vgpr r, lane L -> row = r + 8*(L>>4), col = L&15
    const int rowbase = m0 + half * 8;
#pragma unroll
    for (int r = 0; r < 8; ++r) {
        float* out = H + (size_t)(rowbase + r) * NHID + n0 + l15;
        out[0]  = acc0[r];
        out[16] = acc1[r];
        out[32] = acc2[r];
        out[48] = acc3[r];
    }
}

// =====================================================================
// SpMM d=256: agg1[dst] += w * h[src]   (one wave per edge, 8 coalesced
// global_atomic_add_f32 iterations; h and agg1 are L2-resident: 51.2MB each)
// =====================================================================
__global__ __launch_bounds__(256)
void gcn_spmm256(const int* __restrict__ esrc,
                 const int* __restrict__ edst,
                 const float* __restrict__ ew,
                 const float* __restrict__ H,
                 float* __restrict__ AGG) {
    const int edge = blockIdx.x * 8 + (threadIdx.x >> 5);
    const int lane = threadIdx.x & 31;
    if (edge >= NEDGES) return;
    const int s = esrc[edge];
    const int d = edst[edge];
    const float w = ew[edge];
    const float* hs = H + (size_t)s * NHID;
    float* od = AGG + (size_t)d * NHID;
#pragma unroll
    for (int f = 0; f < NHID; f += 32) {
        unsafeAtomicAdd(&od[f + lane], w * hs[f + lane]);
    }
}

// =====================================================================
// GEMM 2: z = relu(agg1 + b1) @ W2   [50000 x 256] * [256 x 40]
// bias + relu fused into A-fragment loads; N padded to 48 (3 tiles) with
// guards. One wave per 16-row tile, 3 accumulators, 192 wmma / wave.
// =====================================================================
__global__ __launch_bounds__(32)
void gcn_gemm2(const float* __restrict__ AGG1,
               const float* __restrict__ b1,
               const float* __restrict__ W2,
               float* __restrict__ Z) {
    const int lane = threadIdx.x;
    const int m0   = blockIdx.x * 16;
    const int half = lane >> 4;
    const int kh   = half * 2;
    const int l15  = lane & 15;
    const int mrow = m0 + l15;

    v8f acc0 = {}, acc1 = {}, acc2 = {};
    const float* arow = AGG1 + (size_t)mrow * NHID;

    for (int k = 0; k < NHID; k += 4) {
        v2f a;
        a.x = fmaxf(arow[k + kh]     + b1[k + kh],     0.0f);   // fused +b1, relu
        a.y = fmaxf(arow[k + kh + 1] + b1[k + kh + 1], 0.0f);

        const float* brow0 = W2 + (size_t)(k + kh) * NCLS;
        const float* brow1 = brow0 + NCLS;

        v2f b;
        // tile 0: cols 0..15 (always in range)
        b.x = brow0[l15];
        b.y = brow1[l15];
        acc0 = __builtin_amdgcn_wmma_f32_16x16x4_f32(false, a, false, b, (short)0, acc0, false, false);
        // tile 1: cols 16..31 (always in range)
        b.x = brow0[16 + l15];
        b.y = brow1[16 + l15];
        acc1 = __builtin_amdgcn_wmma_f32_16x16x4_f32(false, a, false, b, (short)0, acc1, false, false);
        // tile 2: cols 32..47 -> only 32..39 valid
        const int c2 = 32 + l15;
        b.x = (c2 < NCLS) ? brow0[c2] : 0.0f;
        b.y = (c2 < NCLS) ? brow1[c2] : 0.0f;
        acc2 = __builtin_amdgcn_wmma_f32_16x16x4_f32(false, a, false, b, (short)0, acc2, false, false);
    }

    const int rowbase = m0 + half * 8;
#pragma unroll
    for (int r = 0; r < 8; ++r) {
        float* out = Z + (size_t)(rowbase + r) * NCLS + l15;
        out[0]  = acc0[r];
        out[16] = acc1[r];
        if (32 + l15 < NCLS) out[32] = acc2[r];
    }
}

// =====================================================================
// SpMM d=40: agg2[dst] += w * z[src]   (one wave per edge; z/agg2 = 8MB each,
// fully L2-resident)
// =====================================================================
__global__ __launch_bounds__(256)
void gcn_spmm40(const int* __restrict__ esrc,
                const int* __restrict__ edst,
                const float* __restrict__ ew,
                const float* __restrict__ Z,
                float* __restrict__ AGG) {
    const int edge = blockIdx.x * 8 + (threadIdx.x >> 5);
    const int lane = threadIdx.x & 31;
    if (edge >= NEDGES) return;
    const int s = esrc[edge];
    const int d = edst[edge];
    const float w = ew[edge];
    const float* zs = Z + (size_t)s * NCLS;
    float* od = AGG + (size_t)d * NCLS;
    if (lane < NCLS - 32) {  // lanes 0..7 carry the tail element 32..39
        unsafeAtomicAdd(&od[32 + lane], w * zs[32 + lane]);
    }
    unsafeAtomicAdd(&od[lane], w * zs[lane]);
}

// =====================================================================
// Epilogue: out = log_softmax(agg2 + b2) per row of 40 (one wave per node,
// wave32 shuffle reductions)
// =====================================================================
__global__ __launch_bounds__(256)
void gcn_logsoftmax(const float* __restrict__ AGG2,
                    const float* __restrict__ b2,
                    float* __restrict__ OUT) {
    const int node = blockIdx.x * 8 + (threadIdx.x >> 5);
    const int lane = threadIdx.x & 31;
    if (node >= NNODES) return;

    const float* row = AGG2 + (size_t)node * NCLS;
    const bool has2 = (lane + 32) < NCLS;  // lanes 0..7

    float v0 = row[lane] + b2[lane];
    float v1 = has2 ? (row[lane + 32] + b2[lane + 32]) : -__builtin_inff();

    // max over 40
    float m = fmaxf(v0, v1);
#pragma unroll
    for (int off = 16; off >= 1; off >>= 1)
        m = fmaxf(m, __shfl_xor(m, off, 32));

    // sum of exp
    float s = __expf(v0 - m) + (has2 ? __expf(v1 - m) : 0.0f);
#pragma unroll
    for (int off = 16; off >= 1; off >>= 1)
        s += __shfl_xor(s, off, 32);

    const float lse = m + __logf(s);
    float* orow = OUT + (size_t)node * NCLS;
    orow[lane] = v0 - lse;
    if (has2) orow[lane + 32] = v1 - lse;
}

// =====================================================================
// Launcher
// inputs: 0:x 1:edge_src 2:edge_dst 3:edge_w 4:W1 5:b1 6:W2 7:b2
// =====================================================================
extern "C" void kernel_launch(void* const* d_in, const int* in_sizes, int n_in,
                              void* d_out, int out_size, void* d_ws, size_t ws_size,
                              hipStream_t stream) {
    const float* x    = (const float*)d_in[0];
    const int*   esrc = (const int*)  d_in[1];
    const int*   edst = (const int*)  d_in[2];
    const float* ew   = (const float*)d_in[3];
    const float* W1   = (const float*)d_in[4];
    const float* b1   = (const float*)d_in[5];
    const float* W2   = (const float*)d_in[6];
    const float* b2   = (const float*)d_in[7];

    // workspace layout (fp32): h | agg1 | z | agg2  (total ~118.4 MB)
    float* h    = (float*)d_ws;
    float* agg1 = h    + (size_t)NNODES * NHID;
    float* z    = agg1 + (size_t)NNODES * NHID;
    float* agg2 = z    + (size_t)NNODES * NCLS;

    // accumulators must start at zero every call (graph-capture safe)
    hipMemsetAsync(agg1, 0, (size_t)NNODES * NHID * sizeof(float), stream);
    hipMemsetAsync(agg2, 0, (size_t)NNODES * NCLS * sizeof(float), stream);

    // 1) h = X @ W1
    gcn_gemm1<<<NNODES / 16, dim3(32, 4), 0, stream>>>(x, W1, h);
    // 2) agg1 = scatter_add(w * h[src] -> dst)
    gcn_spmm256<<<NEDGES / 8, 256, 0, stream>>>(esrc, edst, ew, h, agg1);
    // 3) z = relu(agg1 + b1) @ W2
    gcn_gemm2<<<NNODES / 16, 32, 0, stream>>>(agg1, b1, W2, z);
    // 4) agg2 = scatter_add(w * z[src] -> dst)
    gcn_spmm40<<<NEDGES / 8, 256, 0, stream>>>(esrc, edst, ew, z, agg2);
    // 5) out = log_softmax(agg2 + b2)
    gcn_logsoftmax<<<NNODES / 8, 256, 0, stream>>>(agg2, b2, (float*)d_out);
}